// DeformConvV2_88175678587670
// MI455X (gfx1250) — compile-verified
//
#include <hip/hip_runtime.h>
#include <math.h>

typedef float v2f __attribute__((ext_vector_type(2)));
typedef float v8f __attribute__((ext_vector_type(8)));
typedef unsigned int u32x4 __attribute__((ext_vector_type(4)));
typedef int i32x4 __attribute__((ext_vector_type(4)));
typedef int i32x8 __attribute__((ext_vector_type(8)));

#define BATCH 8
#define CHI 128
#define CHO 128
#define HH 64
#define WW 64
#define KK 9
#define HW (HH * WW)            // 4096
#define NPIX (BATCH * HW)       // 32768
#define IMG_STRIDE (CHI * HW)   // 524288
#define OW_CH 27

// ---------------------------------------------------------------------------
// Kernel 0: repack main-conv weights [o][c][kk] -> wt[kk][o][c] so each
// (kk, cb) weight tile is a dense 2D tile the Tensor Data Mover can DMA.
// ---------------------------------------------------------------------------
__global__ __launch_bounds__(256) void weight_repack_kernel(
    const float* __restrict__ w, float* __restrict__ wt) {
  const int idx = blockIdx.x * 256 + threadIdx.x;  // [kk][o][c]
  const int kk = idx >> 14;          // /16384
  const int r = idx & 16383;
  const int o = r >> 7, c = r & 127;
  wt[idx] = w[(o * CHI + c) * 9 + kk];
}

// ---------------------------------------------------------------------------
// Kernel 1: offset/mask conv. One thread per output pixel, off_w staged
// through LDS in 32-input-channel chunks. om layout: pixel-major [pid][27].
// ---------------------------------------------------------------------------
__global__ __launch_bounds__(256) void offset_conv_kernel(
    const float* __restrict__ x, const float* __restrict__ off_w,
    const float* __restrict__ off_b, float* __restrict__ om) {
  __shared__ float sow[32 * 9 * OW_CH];  // 7776 floats = 31 KB
  const int t = threadIdx.x;
  const int pid = blockIdx.x * 256 + t;
  const int b = pid >> 12;
  const int hw = pid & 4095;
  const int h = hw >> 6, wc = hw & 63;

  float acc[OW_CH];
#pragma unroll
  for (int j = 0; j < OW_CH; ++j) acc[j] = off_b[j];

  const float* xb = x + b * IMG_STRIDE;
  for (int cb = 0; cb < CHI; cb += 32) {
    __syncthreads();
    for (int i = t; i < 32 * 9 * OW_CH; i += 256) {
      int j = i % OW_CH;
      int ck = i / OW_CH;  // cl*9 + k
      int cl = ck / 9, k = ck % 9;
      sow[ck * OW_CH + j] = off_w[j * (CHI * 9) + (cb + cl) * 9 + k];
    }
    __syncthreads();
    for (int cl = 0; cl < 32; ++cl) {
      const float* xc = xb + (cb + cl) * HW;
#pragma unroll
      for (int k = 0; k < 9; ++k) {
        const int ky = k / 3, kx = k % 3;
        const int yy = h - 1 + ky, xx = wc - 1 + kx;
        float xv = 0.f;
        if (yy >= 0 && yy < HH && xx >= 0 && xx < WW) xv = xc[yy * WW + xx];
        const float* wr = &sow[(cl * 9 + k) * OW_CH];
#pragma unroll
        for (int j = 0; j < OW_CH; ++j) acc[j] = fmaf(xv, wr[j], acc[j]);
      }
    }
  }
  float* orow = om + pid * OW_CH;
#pragma unroll
  for (int j = 0; j < OW_CH; ++j) orow[j] = acc[j];
}

// ---------------------------------------------------------------------------
// Kernel 2: fused bilinear sampling + implicit GEMM on v_wmma_f32_16x16x4_f32,
// with weight tiles brought into LDS by the Tensor Data Mover (TENSORcnt)
// overlapping the 4-corner gather work.
// Each workgroup: 64 pixels x 128 output channels, 8 waves.
// ---------------------------------------------------------------------------
#define WK_STR 66  // 64 data floats + 2 pad (TDM pad_interval=64dw, amount=2dw)
#define V_STR 66

// TDM descriptor: 2D tile, 128 rows (o) x 64 floats (c), row stride 128 floats,
// destination LDS rows padded to 66 floats. CDNA5 ISA.
__device__ __forceinline__ void tdm_load_weight_tile(const float* gsrc,
                                                     unsigned lds_off) {
  const unsigned long long ga = (unsigned long long)(size_t)gsrc;
  u32x4 g0;
  g0.x = 1u;                                     // count=1 (valid), user mode
  g0.y = lds_off;                                // lds_addr (bytes)
  g0.z = (unsigned)(ga & 0xFFFFFFFFu);           // global_addr[31:0]
  g0.w = (unsigned)((ga >> 32) & 0x1FFFFFFu)     // global_addr[56:32]
         | (2u << 30);                           // type = 2 ("image")
  i32x8 g1;
  g1[0] = (2 << 16)        // data_size = 4 bytes
          | (1 << 20)      // pad_enable
          | (5 << 22)      // pad_interval: 64 DWORDs
          | (1 << 25);     // pad_amount: 2 DWORDs
  g1[1] = (64 << 16);      // tensor_dim0 = 64   (bits 79:48, low half)
  g1[2] = (128 << 16);     // tensor_dim1 = 128  (bits 111:80, low half)
  g1[3] = (64 << 16);      // tile_dim0 = 64     (bits 127:112)
  g1[4] = 128;             // tile_dim1 = 128    (bits 143:128)
  g1[5] = 128;             // tensor_dim0_stride = 128 (bits 207:160, low 32)
  g1[6] = 0;
  g1[7] = 0;
  const i32x4 z4 = {0, 0, 0, 0};                 // groups 2/3 unused (2D tile)
  const i32x8 z8 = {0, 0, 0, 0, 0, 0, 0, 0};
  __builtin_amdgcn_tensor_load_to_lds(g0, g1, z4, z4, z8, 0);
}

__global__ __launch_bounds__(256) void deform_gemm_kernel(
    const float* __restrict__ x, const float* __restrict__ wt,
    const float* __restrict__ bias, const float* __restrict__ om,
    float* __restrict__ y) {
  __shared__ float s_wk[CHO * WK_STR];  // 33 KB, filled by TDM
  __shared__ float s_val[64 * V_STR];   // 16.5 KB
  __shared__ int s_cidx[64 * 4];
  __shared__ float s_cwgt[64 * 4];

  const int t = threadIdx.x;
  const int g = blockIdx.x;  // 512 workgroups, 64 pixels each
  const int lane = t & 31;   // wave32
  const int wv = t >> 5;     // 8 waves
  const int ln = lane & 15;
  const int hi = lane >> 4;  // K-half selector per ISA 16x4 layout
  const int o0 = wv * 16;
  const int b_img = g >> 6;  // 64 WGs per batch image
  const int pid0 = g * 64;
  const float* xb = x + b_img * IMG_STRIDE;
  const unsigned wk_lds = (unsigned)(size_t)(void*)s_wk;  // LDS byte offset

  v8f acc0 = {0.f, 0.f, 0.f, 0.f, 0.f, 0.f, 0.f, 0.f};
  v8f acc1 = acc0, acc2 = acc0, acc3 = acc0;

  // prologue: kick off DMA of first weight tile (kk=0, cb=0)
  if (wv == 0) tdm_load_weight_tile(wt, wk_lds);

  for (int kk = 0; kk < KK; ++kk) {
    __syncthreads();  // previous GEMM done; s_cidx free to rewrite
    if (t < 64) {
      const int p = t;
      const int pid = pid0 + p;
      const int h = (pid & 4095) >> 6;
      const int wc = pid & 63;
      const float* omr = om + pid * OW_CH;
      const float offy = omr[2 * kk], offx = omr[2 * kk + 1];
      const float m = 1.f / (1.f + __expf(-omr[18 + kk]));  // sigmoid mask
      const float py = offy + (float)(h - 1 + kk / 3);
      const float px = offx + (float)(wc - 1 + kk % 3);
      const float fy = floorf(py), fx = floorf(px);
      const float ly = py - fy, lx = px - fx;
      const int y0 = (int)fy, x0 = (int)fx;
      const float wgt[4] = {(1.f - ly) * (1.f - lx), (1.f - ly) * lx,
                            ly * (1.f - lx), ly * lx};
      const int dy[4] = {0, 0, 1, 1}, dx[4] = {0, 1, 0, 1};
#pragma unroll
      for (int j = 0; j < 4; ++j) {
        const int yi = y0 + dy[j], xi = x0 + dx[j];
        const bool ok = (yi >= 0) && (yi < HH) && (xi >= 0) && (xi < WW);
        const int yc = min(max(yi, 0), HH - 1);
        const int xc = min(max(xi, 0), WW - 1);
        s_cidx[p * 4 + j] = yc * WW + xc;
        s_cwgt[p * 4 + j] = ok ? wgt[j] * m : 0.f;  // fold modulation mask
      }
    }
    __syncthreads();  // corner data ready

    for (int cb = 0; cb < CHI; cb += 64) {
      // stage sampled values val[p][c] (4-corner gather from L2-resident x);
      // overlaps the in-flight TDM weight DMA
      {
        const int p = t & 63;
        const int cq = (t >> 6) * 16;
        const int i0 = s_cidx[p * 4 + 0], i1 = s_cidx[p * 4 + 1];
        const int i2 = s_cidx[p * 4 + 2], i3 = s_cidx[p * 4 + 3];
        const float w0 = s_cwgt[p * 4 + 0], w1 = s_cwgt[p * 4 + 1];
        const float w2 = s_cwgt[p * 4 + 2], w3 = s_cwgt[p * 4 + 3];
#pragma unroll 4
        for (int cl = 0; cl < 16; ++cl) {
          const float* xc = xb + (cb + cq + cl) * HW;
          const float v = w0 * xc[i0] + w1 * xc[i1] + w2 * xc[i2] + w3 * xc[i3];
          s_val[p * V_STR + cq + cl] = v;
        }
      }
      if (wv == 0) __builtin_amdgcn_s_wait_tensorcnt(0);  // weight DMA done
      __syncthreads();  // s_val + s_wk ready for everyone

      // GEMM: 16 K-chunks of 4, 4 pixel-tiles per wave
#pragma unroll 4
      for (int cs = 0; cs < 64; cs += 4) {
        const int kofs = cs + 2 * hi;
        v2f a = *(const v2f*)&s_wk[(o0 + ln) * WK_STR + kofs];
        v2f b0 = *(const v2f*)&s_val[(ln)*V_STR + kofs];
        v2f b1 = *(const v2f*)&s_val[(16 + ln) * V_STR + kofs];
        v2f b2 = *(const v2f*)&s_val[(32 + ln) * V_STR + kofs];
        v2f b3 = *(const v2f*)&s_val[(48 + ln) * V_STR + kofs];
        acc0 = __builtin_amdgcn_wmma_f32_16x16x4_f32(false, a, false, b0,
                                                     (short)0, acc0, false, false);
        acc1 = __builtin_amdgcn_wmma_f32_16x16x4_f32(false, a, false, b1,
                                                     (short)0, acc1, false, false);
        acc2 = __builtin_amdgcn_wmma_f32_16x16x4_f32(false, a, false, b2,
                                                     (short)0, acc2, false, false);
        acc3 = __builtin_amdgcn_wmma_f32_16x16x4_f32(false, a, false, b3,
                                                     (short)0, acc3, false, false);
      }
      __syncthreads();  // GEMM done; s_wk/s_val free

      // launch DMA for the next weight tile (if any): runs during the next
      // phase's gather staging
      if (wv == 0) {
        const bool last = (kk == KK - 1) && (cb == 64);
        if (!last) {
          const int nkk = (cb == 0) ? kk : kk + 1;
          const int ncb = (cb == 0) ? 64 : 0;
          tdm_load_weight_tile(wt + nkk * (CHO * CHI) + ncb, wk_lds);
        }
      }
    }
  }

  // epilogue: C/D layout -> lane (r, hi) holds row o0 + 8*hi + r, col ln
#pragma unroll
  for (int r = 0; r < 8; ++r) {
    const int o = o0 + 8 * hi + r;
    const float bb = bias[o];
    y[o * NPIX + pid0 + 0 * 16 + ln] = acc0[r] + bb;
    y[o * NPIX + pid0 + 1 * 16 + ln] = acc1[r] + bb;
    y[o * NPIX + pid0 + 2 * 16 + ln] = acc2[r] + bb;
    y[o * NPIX + pid0 + 3 * 16 + ln] = acc3[r] + bb;
  }
}

// ---------------------------------------------------------------------------
// Kernel 3: per-channel batch statistics (population variance, eps=1e-5).
// ---------------------------------------------------------------------------
__global__ __launch_bounds__(256) void stats_kernel(const float* __restrict__ y,
                                                    float* __restrict__ mean,
                                                    float* __restrict__ rstd) {
  __shared__ float ss[256];
  __shared__ float sq[256];
  const int o = blockIdx.x;
  const int t = threadIdx.x;
  float s = 0.f, q = 0.f;
  for (int i = t; i < NPIX; i += 256) {
    const float v = y[o * NPIX + i];
    s += v;
    q += v * v;
  }
  ss[t] = s;
  sq[t] = q;
  __syncthreads();
  for (int st = 128; st > 0; st >>= 1) {
    if (t < st) {
      ss[t] += ss[t + st];
      sq[t] += sq[t + st];
    }
    __syncthreads();
  }
  if (t == 0) {
    const float inv_n = 1.f / (float)NPIX;
    const float mn = ss[0] * inv_n;
    const float var = sq[0] * inv_n - mn * mn;
    mean[o] = mn;
    rstd[o] = rsqrtf(var + 1e-5f);
  }
}

// ---------------------------------------------------------------------------
// Kernel 4: BN (batch stats) + ReLU, transpose back to NCHW output layout.
// ---------------------------------------------------------------------------
__global__ __launch_bounds__(256) void bn_relu_kernel(
    const float* __restrict__ y, const float* __restrict__ mean,
    const float* __restrict__ rstd, const float* __restrict__ gamma,
    const float* __restrict__ beta, float* __restrict__ out) {
  const int idx = blockIdx.x * 256 + threadIdx.x;  // flat [b][o][hw]
  const int hw = idx & 4095;
  const int o = (idx >> 12) & 127;
  const int b = idx >> 19;
  const float v = y[o * NPIX + b * HW + hw];
  const float r = (v - mean[o]) * rstd[o] * gamma[o] + beta[o];
  out[idx] = fmaxf(r, 0.f);
}

// ---------------------------------------------------------------------------
extern "C" void kernel_launch(void* const* d_in, const int* in_sizes, int n_in,
                              void* d_out, int out_size, void* d_ws,
                              size_t ws_size, hipStream_t stream) {
  (void)in_sizes; (void)n_in; (void)out_size; (void)ws_size;
  const float* x = (const float*)d_in[0];
  const float* off_w = (const float*)d_in[1];
  const float* off_b = (const float*)d_in[2];
  const float* w = (const float*)d_in[3];
  const float* bias = (const float*)d_in[4];
  const float* gamma = (const float*)d_in[5];
  const float* beta = (const float*)d_in[6];

  float* ws = (float*)d_ws;
  float* om = ws;                   // NPIX*27        = 884,736 floats
  float* y = om + NPIX * OW_CH;     // CHO*NPIX       = 4,194,304 floats
  float* mean = y + CHO * NPIX;     // 128
  float* rstd = mean + CHO;         // 128
  float* wt = rstd + CHO;           // KK*CHO*CHI     = 147,456 floats
  float* out = (float*)d_out;

  weight_repack_kernel<<<(KK * CHO * CHI) / 256, 256, 0, stream>>>(w, wt);
  offset_conv_kernel<<<NPIX / 256, 256, 0, stream>>>(x, off_w, off_b, om);
  deform_gemm_kernel<<<NPIX / 64, 256, 0, stream>>>(x, wt, bias, om, y);
  stats_kernel<<<CHO, 256, 0, stream>>>(y, mean, rstd);
  bn_relu_kernel<<<(BATCH * CHO * HW) / 256, 256, 0, stream>>>(y, mean, rstd,
                                                               gamma, beta, out);
}